// LSTMModel_18640158064954
// MI455X (gfx1250) — compile-verified
//
#include <hip/hip_runtime.h>
#include <hip/hip_bf16.h>
#include <math.h>

typedef __attribute__((ext_vector_type(2))) float v2f;
typedef __attribute__((ext_vector_type(8))) float v8f;

#define B_   32
#define S_   256
#define D_   31
#define H_   200
#define L_   32
#define C_   5
#define G4H  800                       // 4*H
#define SA   202                       // LDS row stride for h/c/x (floats, even, conflict-free)
#define SG   804                       // LDS row stride for gate buffer
#define STEP_STRIDE  (B_*H_)           // 6400 floats per timestep slab
#define LAYER_STRIDE (S_*STEP_STRIDE)  // per-layer sequence buffer
#define XW0_STEP     (B_*G4H)          // 25600

// ---------------------------------------------------------------------------
// Kernel A: layer-0 input projection over all timesteps (parallel, K=31)
//   xw0[t][b][j] = x[b,t,:] . W_ih0[j,:] + b_ih0[j] + b_hh0[j]
// ---------------------------------------------------------------------------
__global__ void proj0_kernel(const float* __restrict__ x,
                             const float* __restrict__ Wih0,
                             const float* __restrict__ bih0,
                             const float* __restrict__ bhh0,
                             float* __restrict__ xw0)
{
  int gid = blockIdx.x * blockDim.x + threadIdx.x;
  if (gid >= S_ * B_ * G4H) return;
  int j  = gid % G4H;
  int bt = gid / G4H;
  int b  = bt % B_;
  int t  = bt / B_;
  const float* xr = x    + (size_t)b * S_ * D_ + (size_t)t * D_;
  const float* wr = Wih0 + (size_t)j * D_;
  float acc = bih0[j] + bhh0[j];
#pragma unroll
  for (int d = 0; d < D_; ++d) acc += xr[d] * wr[d];
  xw0[(size_t)t * XW0_STEP + (size_t)b * G4H + j] = acc;
}

// ---------------------------------------------------------------------------
// fp32 WMMA GEMM accumulate: C(2x2 tiles of 16x16) += A(32xK=200) * B(K=200xN=32)
// A fragments come from LDS (h or x state), B fragments stream from L2 weights.
// A 16x4 f32 layout: lane 0-15 -> M rows, VGPR j holds K = 2*(lane>>4)+j.
// B 4x16 f32 layout: lane 0-15 -> N cols, VGPR j holds K = 2*(lane>>4)+j.
// ---------------------------------------------------------------------------
__device__ __forceinline__ void gemm200(const float* __restrict__ A0,
                                        const float* __restrict__ A1,
                                        const float* __restrict__ B0,
                                        const float* __restrict__ B1,
                                        v8f& c00, v8f& c01, v8f& c10, v8f& c11)
{
#pragma unroll 2
  for (int kk = 0; kk < 50; ++kk) {
    v2f a0 = *(const v2f*)(A0 + kk * 4);
    v2f a1 = *(const v2f*)(A1 + kk * 4);
    v2f b0 = *(const v2f*)(B0 + kk * 4);
    v2f b1 = *(const v2f*)(B1 + kk * 4);
    c00 = __builtin_amdgcn_wmma_f32_16x16x4_f32(false, a0, false, b0, (short)0, c00, false, false);
    c01 = __builtin_amdgcn_wmma_f32_16x16x4_f32(false, a0, false, b1, (short)0, c01, false, false);
    c10 = __builtin_amdgcn_wmma_f32_16x16x4_f32(false, a1, false, b0, (short)0, c10, false, false);
    c11 = __builtin_amdgcn_wmma_f32_16x16x4_f32(false, a1, false, b1, (short)0, c11, false, false);
  }
}

// ---------------------------------------------------------------------------
// Kernel B: layer-pipelined wavefront LSTM.
// grid = 32 blocks (one per layer), block = 800 threads = 25 waves.
// Layer l at step t waits (acquire) on layer l-1's progress flag, loads its
// h_t from L2, runs two K=200 fp32-WMMA GEMMs (input proj + recurrence),
// applies gates, publishes h_t (release).
// ---------------------------------------------------------------------------
__global__ void __launch_bounds__(800, 1)
lstm_wavefront_kernel(const float* __restrict__ Whh0,   // (800,200)
                      const float* __restrict__ Wih,    // (L-1,800,200)
                      const float* __restrict__ Whh,    // (L-1,800,200)
                      const float* __restrict__ bih,    // (L-1,800)
                      const float* __restrict__ bhh,    // (L-1,800)
                      const float* __restrict__ xw0,    // (S,B,800)
                      float* __restrict__ seq,          // (L,S,B,H)
                      int* __restrict__ flags)          // (L)
{
  extern __shared__ float smem[];
  float* h_s = smem;                 // 32*SA
  float* c_s = h_s + 32 * SA;        // 32*SA
  float* x_s = c_s + 32 * SA;        // 32*SA
  float* g_s = x_s + 32 * SA;        // 32*SG

  const int l    = blockIdx.x;
  const int tid  = threadIdx.x;
  const int lane = tid & 31;
  const int wave = tid >> 5;         // 0..24

  const float* Wi = (l > 0) ? (Wih + (size_t)(l - 1) * G4H * H_) : nullptr;
  const float* Wh = (l > 0) ? (Whh + (size_t)(l - 1) * G4H * H_) : Whh0;
  const float* bi = (l > 0) ? (bih + (size_t)(l - 1) * G4H) : nullptr;
  const float* bh = (l > 0) ? (bhh + (size_t)(l - 1) * G4H) : nullptr;
  const float* seq_in  = seq + (size_t)(l - 1) * LAYER_STRIDE;
  float*       seq_out = seq + (size_t)l * LAYER_STRIDE;

  // zero h and c state (contiguous in LDS)
  for (int i = tid; i < 2 * 32 * SA; i += 800) h_s[i] = 0.f;
  __syncthreads();

  // per-wave fragment coordinates
  const int mrow  = lane & 15;
  const int ksub  = (lane >> 4) << 1;          // 0 or 2
  const int ncol0 = wave * 32 + (lane & 15);   // first of two 16-wide N tiles
  const int ncol1 = ncol0 + 16;
  const float* A0h = h_s + mrow * SA + ksub;
  const float* A1h = A0h + 16 * SA;
  const float* A0x = x_s + mrow * SA + ksub;
  const float* A1x = A0x + 16 * SA;
  const float* Bh0 = Wh + (size_t)ncol0 * H_ + ksub;
  const float* Bh1 = Wh + (size_t)ncol1 * H_ + ksub;
  const float* Bi0 = (l > 0) ? (Wi + (size_t)ncol0 * H_ + ksub) : nullptr;
  const float* Bi1 = (l > 0) ? (Wi + (size_t)ncol1 * H_ + ksub) : nullptr;

  for (int t = 0; t < S_; ++t) {
    if (l > 0) {
      if (tid == 0) {
        while (__hip_atomic_load(&flags[l - 1], __ATOMIC_ACQUIRE,
                                 __HIP_MEMORY_SCOPE_AGENT) <= t)
          __builtin_amdgcn_s_sleep(2);
      }
      __syncthreads();
      const float* src = seq_in + (size_t)t * STEP_STRIDE;
      for (int i = tid; i < STEP_STRIDE; i += 800)
        x_s[(i / H_) * SA + (i % H_)] = src[i];
      __syncthreads();
    }

    v8f c00 = {}; v8f c01 = {}; v8f c10 = {}; v8f c11 = {};
    if (l > 0) gemm200(A0x, A1x, Bi0, Bi1, c00, c01, c10, c11);  // x_t @ Wi.T
    gemm200(A0h, A1h, Bh0, Bh1, c00, c01, c10, c11);             // h_{t-1} @ Wh.T

    // spill gate pre-activations: C/D layout -> row = 8*(lane>>4)+r, col = lane&15
    {
      const int rbase = (lane >> 4) << 3;
#pragma unroll
      for (int r = 0; r < 8; ++r) {
        int row0 = rbase + r;
        g_s[row0 * SG + ncol0]        = c00[r];
        g_s[row0 * SG + ncol1]        = c01[r];
        g_s[(row0 + 16) * SG + ncol0] = c10[r];
        g_s[(row0 + 16) * SG + ncol1] = c11[r];
      }
    }
    __syncthreads();

    // gate nonlinearity + state update (6400 elements / 800 threads)
    const float* xw_t = xw0 + (size_t)t * XW0_STEP;
#pragma unroll
    for (int e = 0; e < 8; ++e) {
      int idx = e * 800 + tid;     // 0..6399 == b*H + hh
      int b   = idx / H_;
      int hh  = idx - b * H_;
      float ai, af, ag, ao;
      if (l == 0) {   // layer 0: precomputed input projection (biases folded in)
        const float* p = xw_t + (size_t)b * G4H + hh;
        ai = p[0]; af = p[200]; ag = p[400]; ao = p[600];
      } else {
        ai = bi[hh]       + bh[hh];
        af = bi[hh + 200] + bh[hh + 200];
        ag = bi[hh + 400] + bh[hh + 400];
        ao = bi[hh + 600] + bh[hh + 600];
      }
      float gi = g_s[b * SG + hh      ] + ai;
      float gf = g_s[b * SG + hh + 200] + af;
      float gg = g_s[b * SG + hh + 400] + ag;
      float go = g_s[b * SG + hh + 600] + ao;
      float si = 1.f / (1.f + expf(-gi));
      float sf = 1.f / (1.f + expf(-gf));
      float so = 1.f / (1.f + expf(-go));
      float cn = sf * c_s[b * SA + hh] + si * tanhf(gg);
      float hn = so * tanhf(cn);
      c_s[b * SA + hh] = cn;
      h_s[b * SA + hh] = hn;
      seq_out[(size_t)t * STEP_STRIDE + idx] = hn;
    }
    __syncthreads();

    if (tid == 0) {
      __threadfence();
      __hip_atomic_store(&flags[l], t + 1, __ATOMIC_RELEASE,
                         __HIP_MEMORY_SCOPE_AGENT);
    }
  }
}

// ---------------------------------------------------------------------------
// Kernel C: attention + output head, one block per batch row.
// scores[s] = (hn^T W_att) . x_s + hn . b_att  (avoids materializing lin)
// ---------------------------------------------------------------------------
__global__ void attn_out_kernel(const float* __restrict__ seq31, // (S,B,H)
                                const float* __restrict__ Watt,  // (H,H)
                                const float* __restrict__ batt,  // (H)
                                const float* __restrict__ Wout,  // (C,H)
                                const float* __restrict__ bout,  // (C)
                                float* __restrict__ out)         // (B,C)
{
  __shared__ float hn[H_];
  __shared__ float hnw[H_];
  __shared__ float ew[S_];
  __shared__ float red[S_];
  __shared__ float ctx[H_];
  const int b   = blockIdx.x;
  const int tid = threadIdx.x;   // 256 threads == S

  for (int i = tid; i < H_; i += 256)
    hn[i] = seq31[(size_t)(S_ - 1) * STEP_STRIDE + b * H_ + i];
  __syncthreads();

  for (int k = tid; k < H_; k += 256) {
    float a = 0.f;
    for (int h = 0; h < H_; ++h) a += hn[h] * Watt[h * H_ + k];
    hnw[k] = a;
  }
  float hb = 0.f;
  for (int h = 0; h < H_; ++h) hb += hn[h] * batt[h];
  __syncthreads();

  const float* xs = seq31 + (size_t)tid * STEP_STRIDE + b * H_;
  float sc = hb;
  for (int k = 0; k < H_; ++k) sc += xs[k] * hnw[k];

  // softmax over S=256
  red[tid] = sc; __syncthreads();
  for (int o = 128; o > 0; o >>= 1) {
    if (tid < o) red[tid] = fmaxf(red[tid], red[tid + o]);
    __syncthreads();
  }
  float m = red[0]; __syncthreads();
  float e = expf(sc - m);
  ew[tid] = e; red[tid] = e; __syncthreads();
  for (int o = 128; o > 0; o >>= 1) {
    if (tid < o) red[tid] += red[tid + o];
    __syncthreads();
  }
  float inv = 1.f / red[0];
  __syncthreads();

  if (tid < H_) {
    float a = 0.f;
    for (int s = 0; s < S_; ++s)
      a += ew[s] * seq31[(size_t)s * STEP_STRIDE + b * H_ + tid];
    ctx[tid] = a * inv;
  }
  __syncthreads();
  if (tid < C_) {
    float a = bout[tid];
    for (int h = 0; h < H_; ++h) a += ctx[h] * Wout[tid * H_ + h];
    out[b * C_ + tid] = a;
  }
}

// ---------------------------------------------------------------------------
extern "C" void kernel_launch(void* const* d_in, const int* in_sizes, int n_in,
                              void* d_out, int out_size, void* d_ws, size_t ws_size,
                              hipStream_t stream)
{
  const float* x    = (const float*)d_in[0];
  const float* Wih0 = (const float*)d_in[1];
  const float* Whh0 = (const float*)d_in[2];
  const float* bih0 = (const float*)d_in[3];
  const float* bhh0 = (const float*)d_in[4];
  const float* Wih  = (const float*)d_in[5];
  const float* Whh  = (const float*)d_in[6];
  const float* bih  = (const float*)d_in[7];
  const float* bhh  = (const float*)d_in[8];
  const float* Watt = (const float*)d_in[9];
  const float* batt = (const float*)d_in[10];
  const float* Wout = (const float*)d_in[11];
  const float* bout = (const float*)d_in[12];

  // workspace layout: [flags: 256B][xw0: S*B*800 f32][seq: L*S*B*H f32]
  char*  ws    = (char*)d_ws;
  int*   flags = (int*)ws;
  float* xw0   = (float*)(ws + 256);
  float* seq   = xw0 + (size_t)S_ * B_ * G4H;

  hipMemsetAsync(flags, 0, 256, stream);

  {
    int total = S_ * B_ * G4H;
    proj0_kernel<<<dim3((total + 255) / 256), dim3(256), 0, stream>>>(
        x, Wih0, bih0, bhh0, xw0);
  }

  const int ldsBytes = (3 * 32 * SA + 32 * SG) * (int)sizeof(float); // 180,480 B
  hipFuncSetAttribute((const void*)lstm_wavefront_kernel,
                      hipFuncAttributeMaxDynamicSharedMemorySize, ldsBytes);
  lstm_wavefront_kernel<<<dim3(L_), dim3(800), ldsBytes, stream>>>(
      Whh0, Wih, Whh, bih, bhh, xw0, seq, flags);

  const float* seq31 = seq + (size_t)(L_ - 1) * LAYER_STRIDE;
  attn_out_kernel<<<dim3(B_), dim3(256), 0, stream>>>(
      seq31, Watt, batt, Wout, bout, (float*)d_out);
}